// ChamferDistanceL2_split_5248450036649
// MI455X (gfx1250) — compile-verified
//
#include <hip/hip_runtime.h>

// Chamfer distance (squared L2) for B=16, N=M=4096, D=3 on gfx1250.
//
// Strategy: pairwise "distance GEMM" on the matrix core.
//   C_ij = sq2_j - 2 * dot(x_i, y_j)   via V_WMMA_F32_16X16X4_F32 with
//   augmented K=4 operands: A_i = (-2x,-2y,-2z, 1), B_j = (x',y',z', sq2_j).
//   dist1_i = sq1_i + min_j C_ij  (sq1 added in the final reduction kernel).
// Inputs (786KB each) are L2-resident; one batch of the "B side" is staged
// through LDS in 32KB chunks. Min over columns = shfl_xor over 16 lanes.

typedef float v2f __attribute__((ext_vector_type(2)));
typedef float v8f __attribute__((ext_vector_type(8)));

#define CHUNK 2048   // points staged in LDS per pass (2048 * 16B = 32KB)

__global__ __launch_bounds__(256)
void chamfer_min_wmma(const float* __restrict__ xyzA,   // [B, NA, 3] query side
                      const float* __restrict__ xyzB,   // [B, NB, 3] database side
                      float* __restrict__ minC,         // [B, NA]  min_j (sq2_j - 2 x.y)
                      int NA, int NB)
{
    __shared__ float4 spts[CHUNK];          // {x, y, z, ||p||^2}

    const int b    = blockIdx.y;
    const int tid  = threadIdx.x;
    const int wave = tid >> 5;              // 8 waves / block (wave32)
    const int lane = tid & 31;
    const int half = lane >> 4;             // 0 -> (K0,K1), 1 -> (K2,K3)
    const int l16  = lane & 15;

    const int n0 = blockIdx.x * 128 + wave * 16;   // 16 query rows per wave

    // ---- A fragment (constant over the whole loop) ------------------------
    // 16x4 f32 A layout: lane l holds row M = l%16; half 0 = K0,K1; half 1 = K2,K3.
    const float* pa = xyzA + ((size_t)b * NA + n0 + l16) * 3;
    const float ax = pa[0], ay = pa[1], az = pa[2];
    v2f afrag;
    afrag[0] = half ? (-2.0f * az) : (-2.0f * ax);
    afrag[1] = half ? 1.0f         : (-2.0f * ay);

    float minv[8];
    #pragma unroll
    for (int r = 0; r < 8; ++r) minv[r] = 3.0e38f;

    for (int base = 0; base < NB; base += CHUNK) {
        __syncthreads();
        // ---- stage a chunk of the database side into LDS ------------------
        for (int i = tid; i < CHUNK; i += 256) {
            const float* pb = xyzB + ((size_t)b * NB + base + i) * 3;
            const float x = pb[0], y = pb[1], z = pb[2];
            spts[i] = make_float4(x, y, z, x * x + y * y + z * z);
        }
        __syncthreads();

        // ---- sweep 16-wide column tiles ----------------------------------
        #pragma unroll 4
        for (int t = 0; t < CHUNK; t += 16) {
            // B 4x16 f32 layout: lane l holds col N = l%16; half 0 = rows K0,K1
            // -> (x', y'); half 1 = rows K2,K3 -> (z', sq2).
            const float2* p2 = reinterpret_cast<const float2*>(&spts[t + l16]);
            const float2 bv = p2[half];                 // ds_load_b64
            v2f bfrag; bfrag[0] = bv.x; bfrag[1] = bv.y;

            v8f c = {0.f, 0.f, 0.f, 0.f, 0.f, 0.f, 0.f, 0.f};
            c = __builtin_amdgcn_wmma_f32_16x16x4_f32(
                    /*neg_a=*/false, afrag, /*neg_b=*/false, bfrag,
                    /*c_mod=*/(short)0, c, /*reuse_a=*/false, /*reuse_b=*/false);

            #pragma unroll
            for (int r = 0; r < 8; ++r) minv[r] = fminf(minv[r], c[r]);
        }
    }

    // ---- min over the 16 columns held across lanes (stays within halves) --
    #pragma unroll
    for (int r = 0; r < 8; ++r) {
        float v = minv[r];
        v = fminf(v, __shfl_xor(v, 1, 32));
        v = fminf(v, __shfl_xor(v, 2, 32));
        v = fminf(v, __shfl_xor(v, 4, 32));
        v = fminf(v, __shfl_xor(v, 8, 32));
        minv[r] = v;
    }

    // C layout: half 0 VGPR r -> row r, half 1 VGPR r -> row r+8.
    if (l16 == 0) {
        float* outp = minC + (size_t)b * NA + n0 + half * 8;
        #pragma unroll
        for (int r = 0; r < 8; ++r) outp[r] = minv[r];
    }
}

// Deterministic mean of (minC[i] + ||xyz_i||^2) over `count` points.
__global__ __launch_bounds__(256)
void chamfer_finalize(const float* __restrict__ minC,
                      const float* __restrict__ xyz,   // [count, 3]
                      int count, float* __restrict__ out)
{
    __shared__ float red[256];
    float s = 0.0f;
    for (int i = threadIdx.x; i < count; i += 256) {
        const float x = xyz[3 * i + 0];
        const float y = xyz[3 * i + 1];
        const float z = xyz[3 * i + 2];
        s += minC[i] + (x * x + y * y + z * z);
    }
    red[threadIdx.x] = s;
    __syncthreads();
    for (int off = 128; off > 0; off >>= 1) {
        if (threadIdx.x < off) red[threadIdx.x] += red[threadIdx.x + off];
        __syncthreads();
    }
    if (threadIdx.x == 0) *out = red[0] / (float)count;
}

extern "C" void kernel_launch(void* const* d_in, const int* in_sizes, int n_in,
                              void* d_out, int out_size, void* d_ws, size_t ws_size,
                              hipStream_t stream)
{
    const float* xyz1 = (const float*)d_in[0];   // [16, 4096, 3] fp32
    const float* xyz2 = (const float*)d_in[1];   // [16, 4096, 3] fp32
    float* out = (float*)d_out;                  // [2] fp32

    const int B = 16, N = 4096, M = 4096;

    float* ws1 = (float*)d_ws;                   // [B*N] min_j C  (dist1 part)
    float* ws2 = ws1 + (size_t)B * N;            // [B*M] min_i C  (dist2 part)

    // dist1: queries = xyz1, database = xyz2
    chamfer_min_wmma<<<dim3(N / 128, B), dim3(256), 0, stream>>>(xyz1, xyz2, ws1, N, M);
    // dist2: queries = xyz2, database = xyz1
    chamfer_min_wmma<<<dim3(M / 128, B), dim3(256), 0, stream>>>(xyz2, xyz1, ws2, M, N);

    chamfer_finalize<<<1, 256, 0, stream>>>(ws1, xyz1, B * N, out + 0);
    chamfer_finalize<<<1, 256, 0, stream>>>(ws2, xyz2, B * M, out + 1);
}